// GATNetSelectiveResidualsUpdatedLayerNorm_31628139168050
// MI455X (gfx1250) — compile-verified
//
#include <hip/hip_runtime.h>
#include <hip/hip_bf16.h>
#include <math.h>

typedef __attribute__((ext_vector_type(2))) float v2f;
typedef __attribute__((ext_vector_type(8))) float v8f;

#define N_NODES 8192
#define F_CONV  512   // H*D = 2*256
#define NHEADS  2
#define DHEAD   256

// ---------------------------------------------------------------------------
// device helpers
// ---------------------------------------------------------------------------
__device__ __forceinline__ float gelu_exact(float x) {
    // reference uses exact erf GELU
    return 0.5f * x * (1.0f + erff(x * 0.70710678118654752440f));
}

__device__ __forceinline__ void atomicMaxFloat(float* addr, float val) {
    // signed/unsigned int ordering trick; valid with -inf init
    if (val >= 0.0f)
        atomicMax((int*)addr, __float_as_int(val));
    else
        atomicMin((unsigned int*)addr, (unsigned int)__float_as_int(val));
}

// ---------------------------------------------------------------------------
// WMMA f32 GEMM:  C[M,Nc] = A[M,K] @ B[K,Nc] (+ bias[Nc])
// one wave -> one 16x16 tile, V_WMMA_F32_16X16X4_F32, K stepped by 8 with
// two accumulators to hide the WMMA->WMMA RAW latency.
// launch: block(32,4), grid(Nc/64, M/16)
// ---------------------------------------------------------------------------
__global__ void wmma_gemm_f32(const float* __restrict__ A,
                              const float* __restrict__ B,
                              const float* __restrict__ bias,
                              float* __restrict__ C,
                              int M, int K, int Nc) {
    const int lane    = threadIdx.x;                    // 0..31
    const int colTile = blockIdx.x * blockDim.y + threadIdx.y;
    const int rowTile = blockIdx.y;
    const int row0 = rowTile * 16;
    const int col0 = colTile * 16;
    const int half = lane >> 4;                         // 0: K=0,1  1: K=2,3
    const int l    = lane & 15;

    v8f acc0 = {};
    v8f acc1 = {};

    const float* arow = A + (size_t)(row0 + l) * K + half * 2;
    const float* bcol = B + (size_t)(half * 2) * Nc + col0 + l;

    for (int k = 0; k < K; k += 8) {
        v2f a0, b0, a1, b1;
        a0.x = arow[k];     a0.y = arow[k + 1];
        b0.x = bcol[(size_t)k * Nc];
        b0.y = bcol[(size_t)(k + 1) * Nc];
        acc0 = __builtin_amdgcn_wmma_f32_16x16x4_f32(
            false, a0, false, b0, (short)0, acc0, false, false);

        a1.x = arow[k + 4]; a1.y = arow[k + 5];
        b1.x = bcol[(size_t)(k + 4) * Nc];
        b1.y = bcol[(size_t)(k + 5) * Nc];
        acc1 = __builtin_amdgcn_wmma_f32_16x16x4_f32(
            false, a1, false, b1, (short)0, acc1, false, false);
    }

    const int r = row0 + half * 8;
    const int c = col0 + l;
    const float bv = bias ? bias[c] : 0.0f;
#pragma unroll
    for (int i = 0; i < 8; ++i)
        C[(size_t)(r + i) * Nc + c] = acc0[i] + acc1[i] + bv;
}

// ---------------------------------------------------------------------------
// per-(node,head) attention logits: a_src/a_dst = <xw[n,h,:], att[h,:]>
// one wave per (n,h); 8 elems/lane + shuffle reduction
// ---------------------------------------------------------------------------
__global__ void node_attn_kernel(const float* __restrict__ xw,
                                 const float* __restrict__ att_src,
                                 const float* __restrict__ att_dst,
                                 float* __restrict__ a_src,
                                 float* __restrict__ a_dst, int nNodes) {
    int gid  = blockIdx.x * blockDim.x + threadIdx.x;
    int wave = gid >> 5, lane = gid & 31;
    if (wave >= nNodes * NHEADS) return;
    int n = wave >> 1, h = wave & 1;
    const float* xr = xw + (size_t)n * F_CONV + h * DHEAD;
    const float* as = att_src + h * DHEAD;
    const float* ad = att_dst + h * DHEAD;
    float ss = 0.0f, sd = 0.0f;
    for (int i = lane; i < DHEAD; i += 32) {
        float xv = xr[i];
        ss += xv * as[i];
        sd += xv * ad[i];
    }
    for (int o = 16; o > 0; o >>= 1) {
        ss += __shfl_down(ss, o, 32);
        sd += __shfl_down(sd, o, 32);
    }
    if (lane == 0) {
        a_src[n * NHEADS + h] = ss;
        a_dst[n * NHEADS + h] = sd;
    }
}

// ---------------------------------------------------------------------------
// init: hbuf = 0, emax = -inf, denom = 0
// ---------------------------------------------------------------------------
__global__ void init_kernel(float* hbuf, float* emax, float* denom,
                            int total, int n2) {
    int i = blockIdx.x * blockDim.x + threadIdx.x;
    if (i < total) hbuf[i] = 0.0f;
    if (i < n2) { emax[i] = -__builtin_inff(); denom[i] = 0.0f; }
}

// ---------------------------------------------------------------------------
// edge pass 1: segment max of leaky_relu(a_src[s]+a_dst[d]) over dst
// ---------------------------------------------------------------------------
__global__ void edge_max_kernel(const int* __restrict__ ei,
                                const float* __restrict__ a_src,
                                const float* __restrict__ a_dst,
                                float* __restrict__ emax, int E, int nNodes) {
    int e = blockIdx.x * blockDim.x + threadIdx.x;
    if (e >= E + nNodes) return;
    int s, d;
    if (e < E) { s = ei[e]; d = ei[E + e]; } else { s = d = e - E; }
#pragma unroll
    for (int h = 0; h < NHEADS; ++h) {
        float v = a_src[s * NHEADS + h] + a_dst[d * NHEADS + h];
        v = v > 0.0f ? v : 0.2f * v;
        atomicMaxFloat(&emax[d * NHEADS + h], v);
    }
}

// edge pass 2: denom[d] += exp(e - emax[d])
__global__ void edge_sum_kernel(const int* __restrict__ ei,
                                const float* __restrict__ a_src,
                                const float* __restrict__ a_dst,
                                const float* __restrict__ emax,
                                float* __restrict__ denom, int E, int nNodes) {
    int e = blockIdx.x * blockDim.x + threadIdx.x;
    if (e >= E + nNodes) return;
    int s, d;
    if (e < E) { s = ei[e]; d = ei[E + e]; } else { s = d = e - E; }
#pragma unroll
    for (int h = 0; h < NHEADS; ++h) {
        float v = a_src[s * NHEADS + h] + a_dst[d * NHEADS + h];
        v = v > 0.0f ? v : 0.2f * v;
        atomicAdd(&denom[d * NHEADS + h], expf(v - emax[d * NHEADS + h]));
    }
}

// edge pass 3: hbuf[d,:] += alpha * xw[s,:]   (one wave per edge, 16 elems/lane)
__global__ void edge_scatter_kernel(const int* __restrict__ ei,
                                    const float* __restrict__ a_src,
                                    const float* __restrict__ a_dst,
                                    const float* __restrict__ emax,
                                    const float* __restrict__ denom,
                                    const float* __restrict__ xw,
                                    float* __restrict__ hbuf, int E, int nNodes) {
    int gid  = blockIdx.x * blockDim.x + threadIdx.x;
    int wave = gid >> 5, lane = gid & 31;
    if (wave >= E + nNodes) return;
    int s, d;
    if (wave < E) { s = ei[wave]; d = ei[E + wave]; } else { s = d = wave - E; }
    float al[NHEADS];
#pragma unroll
    for (int h = 0; h < NHEADS; ++h) {
        float v = a_src[s * NHEADS + h] + a_dst[d * NHEADS + h];
        v = v > 0.0f ? v : 0.2f * v;
        al[h] = expf(v - emax[d * NHEADS + h]) / denom[d * NHEADS + h];
    }
    const float* xs = xw + (size_t)s * F_CONV;
    float* hd = hbuf + (size_t)d * F_CONV;
    for (int f = lane; f < F_CONV; f += 32)
        atomicAdd(&hd[f], al[f >> 8] * xs[f]);  // f>>8 selects head (D=256)
}

// ---------------------------------------------------------------------------
// h = gelu(hbuf + conv_b)
// ---------------------------------------------------------------------------
__global__ void bias_gelu_kernel(float* __restrict__ h,
                                 const float* __restrict__ b, int total) {
    int i = blockIdx.x * blockDim.x + threadIdx.x;
    if (i >= total) return;
    h[i] = gelu_exact(h[i] + b[i & (F_CONV - 1)]);
}

// ---------------------------------------------------------------------------
// row chain: LN(g1,b1) -> gelu [-> +res] [-> LN(g2,b2)], one block per row
// ---------------------------------------------------------------------------
template <int F, bool ADD_RES, bool SECOND_LN>
__global__ void ln_chain_kernel(float* __restrict__ t,
                                const float* __restrict__ res,
                                const float* __restrict__ g1,
                                const float* __restrict__ b1,
                                const float* __restrict__ g2,
                                const float* __restrict__ b2) {
    __shared__ float sh[F];
    const int n = blockIdx.x, tid = threadIdx.x;
    float* row = t + (size_t)n * F;
    float v = row[tid];

    sh[tid] = v; __syncthreads();
    for (int o = F / 2; o > 0; o >>= 1) { if (tid < o) sh[tid] += sh[tid + o]; __syncthreads(); }
    float mu = sh[0] * (1.0f / F); __syncthreads();
    float dv = v - mu;
    sh[tid] = dv * dv; __syncthreads();
    for (int o = F / 2; o > 0; o >>= 1) { if (tid < o) sh[tid] += sh[tid + o]; __syncthreads(); }
    float var = sh[0] * (1.0f / F); __syncthreads();

    float s = gelu_exact(dv * rsqrtf(var + 1e-5f) * g1[tid] + b1[tid]);
    if (ADD_RES) s += res[(size_t)n * F + tid];

    if (SECOND_LN) {
        sh[tid] = s; __syncthreads();
        for (int o = F / 2; o > 0; o >>= 1) { if (tid < o) sh[tid] += sh[tid + o]; __syncthreads(); }
        float mu2 = sh[0] * (1.0f / F); __syncthreads();
        float d2 = s - mu2;
        sh[tid] = d2 * d2; __syncthreads();
        for (int o = F / 2; o > 0; o >>= 1) { if (tid < o) sh[tid] += sh[tid + o]; __syncthreads(); }
        float var2 = sh[0] * (1.0f / F); __syncthreads();
        s = d2 * rsqrtf(var2 + 1e-5f) * g2[tid] + b2[tid];
    }
    row[tid] = s;
}

// ---------------------------------------------------------------------------
// y = h64 @ W3 + b3  (K=64, Nc=3 — VALU; pad y to stride 4, also emit |y|^2)
// ---------------------------------------------------------------------------
__global__ void head_y_kernel(const float* __restrict__ h64,
                              const float* __restrict__ W3,
                              const float* __restrict__ b3,
                              float* __restrict__ y, float* __restrict__ sq,
                              int nNodes) {
    int n = blockIdx.x * blockDim.x + threadIdx.x;
    if (n >= nNodes) return;
    const float* hr = h64 + (size_t)n * 64;
    float y0 = b3[0], y1 = b3[1], y2 = b3[2];
#pragma unroll 8
    for (int k = 0; k < 64; ++k) {
        float hv = hr[k];
        y0 = fmaf(hv, W3[k * 3 + 0], y0);
        y1 = fmaf(hv, W3[k * 3 + 1], y1);
        y2 = fmaf(hv, W3[k * 3 + 2], y2);
    }
    y[n * 4 + 0] = y0; y[n * 4 + 1] = y1; y[n * 4 + 2] = y2; y[n * 4 + 3] = 0.0f;
    sq[n] = y0 * y0 + y1 * y1 + y2 * y2;
}

// ---------------------------------------------------------------------------
// cdist: out[i,j] = sqrt(max(sq[i]+sq[j]-2<y_i,y_j>,0))  (store-bandwidth bound)
// ---------------------------------------------------------------------------
__global__ void cdist_kernel(const float* __restrict__ y,
                             const float* __restrict__ sq,
                             float* __restrict__ out, int nNodes) {
    int j = blockIdx.x * blockDim.x + threadIdx.x;
    int i = blockIdx.y;
    float yi0 = y[i * 4 + 0], yi1 = y[i * 4 + 1], yi2 = y[i * 4 + 2];
    float sqi = sq[i];
    float4 yj = ((const float4*)y)[j];
    float d2 = sqi + sq[j] - 2.0f * (yi0 * yj.x + yi1 * yj.y + yi2 * yj.z);
    out[(size_t)i * nNodes + j] = d2 > 0.0f ? sqrtf(d2) : 0.0f;
}

// ---------------------------------------------------------------------------
// launch
// ---------------------------------------------------------------------------
extern "C" void kernel_launch(void* const* d_in, const int* in_sizes, int n_in,
                              void* d_out, int out_size, void* d_ws, size_t ws_size,
                              hipStream_t stream) {
    (void)n_in; (void)out_size; (void)ws_size;
    const float* x       = (const float*)d_in[0];
    const int*   ei      = (const int*)  d_in[1];
    const float* conv_W  = (const float*)d_in[2];
    const float* att_src = (const float*)d_in[3];
    const float* att_dst = (const float*)d_in[4];
    const float* conv_b  = (const float*)d_in[5];
    const float* Wa    = (const float*)d_in[6];  const float* ba    = (const float*)d_in[7];
    const float* ga    = (const float*)d_in[8];  const float* bna   = (const float*)d_in[9];
    const float* Wal_a = (const float*)d_in[10]; const float* bal_a = (const float*)d_in[11];
    const float* gra   = (const float*)d_in[12]; const float* bra   = (const float*)d_in[13];
    const float* W1    = (const float*)d_in[14]; const float* b1    = (const float*)d_in[15];
    const float* g1    = (const float*)d_in[16]; const float* bn1   = (const float*)d_in[17];
    const float* Wal1  = (const float*)d_in[18]; const float* bal1  = (const float*)d_in[19];
    const float* gr1   = (const float*)d_in[20]; const float* br1   = (const float*)d_in[21];
    const float* W2    = (const float*)d_in[22]; const float* b2    = (const float*)d_in[23];
    const float* g2    = (const float*)d_in[24]; const float* bn2   = (const float*)d_in[25];
    const float* W3    = (const float*)d_in[26]; const float* b3    = (const float*)d_in[27];

    const int N = in_sizes[0] / 512;   // 8192
    const int E = in_sizes[1] / 2;     // 262144
    const int EN = E + N;

    float* ws    = (float*)d_ws;
    float* xw    = ws;                         // N*512
    float* hbuf  = xw    + (size_t)N * 512;    // N*512
    float* a_s   = hbuf  + (size_t)N * 512;    // N*2
    float* a_d   = a_s   + (size_t)N * 2;
    float* emax  = a_d   + (size_t)N * 2;
    float* denom = emax  + (size_t)N * 2;
    float* t256a = denom + (size_t)N * 2;      // N*256
    float* t256b = t256a + (size_t)N * 256;    // N*256
    float* t128a = t256b + (size_t)N * 256;    // N*128
    float* t128b = t128a + (size_t)N * 128;    // N*128
    float* t64   = t128b + (size_t)N * 128;    // N*64
    float* ybuf  = t64   + (size_t)N * 64;     // N*4
    float* sqbuf = ybuf  + (size_t)N * 4;      // N

    dim3 wblk(32, 4);

    // 1) xw = x @ conv_W           [8192,512] x [512,512]
    wmma_gemm_f32<<<dim3(512 / 64, N / 16), wblk, 0, stream>>>(
        x, conv_W, nullptr, xw, N, 512, 512);

    // 2) attention logits
    node_attn_kernel<<<(N * 2 * 32 + 255) / 256, 256, 0, stream>>>(
        xw, att_src, att_dst, a_s, a_d, N);

    // 3) init accumulators
    init_kernel<<<((size_t)N * 512 + 255) / 256, 256, 0, stream>>>(
        hbuf, emax, denom, N * 512, N * 2);

    // 4-6) segment softmax + message aggregation
    edge_max_kernel<<<(EN + 255) / 256, 256, 0, stream>>>(ei, a_s, a_d, emax, E, N);
    edge_sum_kernel<<<(EN + 255) / 256, 256, 0, stream>>>(ei, a_s, a_d, emax, denom, E, N);
    edge_scatter_kernel<<<((size_t)EN * 32 + 255) / 256, 256, 0, stream>>>(
        ei, a_s, a_d, emax, denom, xw, hbuf, E, N);

    // 7) h = gelu(conv_out + conv_b)
    bias_gelu_kernel<<<((size_t)N * 512 + 255) / 256, 256, 0, stream>>>(
        hbuf, conv_b, N * 512);

    // 8-10) 512 -> 256 block (residual projection + main + LN/gelu/add/LN)
    wmma_gemm_f32<<<dim3(256 / 64, N / 16), wblk, 0, stream>>>(
        hbuf, Wal_a, bal_a, t256b, N, 512, 256);
    wmma_gemm_f32<<<dim3(256 / 64, N / 16), wblk, 0, stream>>>(
        hbuf, Wa, ba, t256a, N, 512, 256);
    ln_chain_kernel<256, true, true><<<N, 256, 0, stream>>>(
        t256a, t256b, ga, bna, gra, bra);

    // 11-13) 256 -> 128 block
    wmma_gemm_f32<<<dim3(128 / 64, N / 16), wblk, 0, stream>>>(
        t256a, Wal1, bal1, t128b, N, 256, 128);
    wmma_gemm_f32<<<dim3(128 / 64, N / 16), wblk, 0, stream>>>(
        t256a, W1, b1, t128a, N, 256, 128);
    ln_chain_kernel<128, true, true><<<N, 128, 0, stream>>>(
        t128a, t128b, g1, bn1, gr1, br1);

    // 14-15) 128 -> 64 block
    wmma_gemm_f32<<<dim3(64 / 64, N / 16), wblk, 0, stream>>>(
        t128a, W2, b2, t64, N, 128, 64);
    ln_chain_kernel<64, false, false><<<N, 64, 0, stream>>>(
        t64, nullptr, g2, bn2, nullptr, nullptr);

    // 16) y head + squared norms
    head_y_kernel<<<(N + 255) / 256, 256, 0, stream>>>(t64, W3, b3, ybuf, sqbuf, N);

    // 17) pairwise distances (256 MB coalesced store)
    cdist_kernel<<<dim3(N / 256, N), 256, 0, stream>>>(ybuf, sqbuf, (float*)d_out, N);
}